// WOS_72842645340328
// MI455X (gfx1250) — compile-verified
//
#include <hip/hip_runtime.h>
#include <hip/hip_bf16.h>
#include <stdint.h>

// Problem constants (fixed by the reference).
#define KK   3
#define CC   3
#define HH   160
#define WW   160
#define BB   4
#define NCH  8
#define DD   27          // C*K*K
#define MDIM 54          // 2*DD
#define HO   158
#define WO   158
#define LL   (HO * WO)   // 24964
#define NTOT (BB * LL)   // 99856
#define TPB  256
#define MXSTRIDE 55      // odd dword stride -> LDS bank-conflict free for wave32

typedef unsigned int v4u __attribute__((ext_vector_type(4)));
typedef int          v4i __attribute__((ext_vector_type(4)));
typedef int          v8i __attribute__((ext_vector_type(8)));

#if __has_builtin(__builtin_amdgcn_tensor_load_to_lds)
#define HAVE_TDM 1
#else
#define HAVE_TDM 0
#endif

#if HAVE_TDM
// 1-D (single row) tensor DMA: nd dwords from global gaddr -> LDS byte offset lds_byte.
// D# packing per CDNA5 ISA:
//  g0: count=1 | lds_addr | global_addr[31:0] | global_addr[56:32] + type=2
//  g1: data_size=2 (4B); tensor_dim0=nd; tensor_dim1=1; tile_dim0=nd; tile_dim1/2=0;
//      tensor_dim0_stride=nd; no padding / iterate / multicast / atomic-barrier.
__device__ __forceinline__ void tdm_load_1d(uint32_t lds_byte, uint64_t gaddr, uint32_t nd) {
  v4u g0;
  g0[0] = 1u;                                                  // count=1, user mode
  g0[1] = lds_byte;                                            // lds_addr
  g0[2] = (uint32_t)gaddr;                                     // global_addr[31:0]
  g0[3] = (uint32_t)((gaddr >> 32) & 0x1FFFFFFull) | (2u << 30); // [56:32] | type=2
  v8i g1;
  g1[0] = (int)(2u << 16);                       // workgroup_mask=0, data_size=4B
  g1[1] = (int)((nd & 0xFFFFu) << 16);           // tensor_dim0[15:0] (bits 63:48)
  g1[2] = (int)((nd >> 16) | (1u << 16));        // tensor_dim0[31:16] | tensor_dim1=1
  g1[3] = (int)((nd & 0xFFFFu) << 16);           // tensor_dim1 hi=0 | tile_dim0=nd
  g1[4] = 0;                                     // tile_dim1=0, tile_dim2=0 (1-D tile)
  g1[5] = (int)nd;                               // tensor_dim0_stride[31:0]
  g1[6] = 0;                                     // stride0 hi | stride1 lo
  g1[7] = 0;                                     // stride1 hi
  v4i g2 = {1, 1, 0, 0};                         // tensor_dim2=1, tensor_dim3=1
  v4i g3 = {0, 0, 0, 0};
#if defined(__clang_major__) && (__clang_major__ >= 23)
  v8i g4 = {0, 0, 0, 0, 0, 0, 0, 0};
  __builtin_amdgcn_tensor_load_to_lds(g0, g1, g2, g3, g4, 0);
#else
  __builtin_amdgcn_tensor_load_to_lds(g0, g1, g2, g3, 0);
#endif
}
#endif

__global__ __launch_bounds__(TPB) void WOS_72842645340328_kernel(
    const float* __restrict__ x, const float* __restrict__ weight,
    const float* __restrict__ bias, const float* __restrict__ mask,
    float* __restrict__ out)
{
  __shared__ float s_w[NCH * MDIM];           // 1.7 KB
  __shared__ float s_m[NCH * MDIM];           // 1.7 KB
  __shared__ float s_b[NCH];
  __shared__ float s_mx[TPB * MXSTRIDE];      // 55 KB: per-thread mx for dynamic index

  const int t = threadIdx.x;

#if HAVE_TDM
  if (t < 32) {  // wave 0 only issues the DMA (wave-level instruction)
    tdm_load_1d((uint32_t)(uintptr_t)&s_w[0], (uint64_t)(uintptr_t)weight, NCH * MDIM);
    tdm_load_1d((uint32_t)(uintptr_t)&s_m[0], (uint64_t)(uintptr_t)mask,   NCH * MDIM);
#if __has_builtin(__builtin_amdgcn_s_wait_tensorcnt)
    __builtin_amdgcn_s_wait_tensorcnt(0);
#endif
  }
#else
  for (int i = t; i < NCH * MDIM; i += TPB) { s_w[i] = weight[i]; s_m[i] = mask[i]; }
#endif
  if (t < NCH) s_b[t] = bias[t];
  __syncthreads();

  const int tid = blockIdx.x * TPB + t;
  if (tid >= NTOT) return;

  const int b  = tid / LL;
  const int l  = tid - b * LL;
  const int ho = l / WO;
  const int wo = l - ho * WO;

  // 3x3x3 unfold window, feature index = c*9 + i*3 + j (channel slowest).
  float v[DD];
  {
    const float* xb = x + (size_t)b * CC * HH * WW;
    #pragma unroll
    for (int c = 0; c < CC; ++c)
      #pragma unroll
      for (int i = 0; i < KK; ++i)
        #pragma unroll
        for (int j = 0; j < KK; ++j)
          v[c * 9 + i * 3 + j] = xb[((size_t)c * HH + ho + i) * WW + wo + j];
  }

  float* myMx = &s_mx[t * MXSTRIDE];
  const float INF = __builtin_inff();

  #pragma unroll 1
  for (int nc = 0; nc < NCH; ++nc) {
    // Weights are wave-uniform: pin them in SGPRs via readfirstlane so the
    // hot cndmask reads a scalar operand and VGPR pressure stays low.
    float w[MDIM];
    #pragma unroll
    for (int u = 0; u < MDIM; ++u)
      w[u] = __uint_as_float(__builtin_amdgcn_readfirstlane(__float_as_uint(s_w[nc * MDIM + u])));
    const float bia =
        __uint_as_float(__builtin_amdgcn_readfirstlane(__float_as_uint(s_b[nc])));

    // mx[u] = (u<27 ? v[u] : -v[u-27]) + mask[nc][u]; keep in VGPRs (static index)
    // and mirror to LDS for the dynamically-indexed read in the e-loop.
    float mx[MDIM];
    float gmax = -INF;
    #pragma unroll
    for (int u = 0; u < MDIM; ++u) {
      const float base = (u < DD) ? v[u] : -v[u - DD];
      const float m = base + s_m[nc * MDIM + u];
      mx[u] = m;
      myMx[u] = m;
      gmax = fmaxf(gmax, m);
    }

    // Rank-sum selection: S_e = sum_u w_u * [mx_u >= mx_e]  (== cumsum of sorted
    // weights at e's descending-sort position, ties measure-zero).
    // Answer = min{ mx_e : S_e <= bias }, else max (li clipped to 0 in reference).
    float best = INF;
    #pragma unroll 1
    for (int e = 0; e < MDIM; ++e) {
      const float mxe = myMx[e];
      float S = 0.0f;
      #pragma unroll
      for (int u = 0; u < MDIM; ++u)
        S += (mx[u] >= mxe) ? w[u] : 0.0f;     // v_cmp + v_cndmask(sgpr) + v_add
      best = ((S <= bia) && (mxe < best)) ? mxe : best;
    }

    // Raw-view output layout: flat index = n*NC + nc.
    out[(size_t)tid * NCH + nc] = (best < INF) ? best : gmax;
  }
}

extern "C" void kernel_launch(void* const* d_in, const int* in_sizes, int n_in,
                              void* d_out, int out_size, void* d_ws, size_t ws_size,
                              hipStream_t stream) {
  (void)in_sizes; (void)n_in; (void)out_size; (void)d_ws; (void)ws_size;
  const float* x      = (const float*)d_in[0];   // [4,3,160,160]
  const float* weight = (const float*)d_in[1];   // [8,54]
  const float* bias   = (const float*)d_in[2];   // [8,1]
  const float* mask   = (const float*)d_in[3];   // [8,54]
  float* out = (float*)d_out;                    // [N*8] raw view of [4,8,158,158]

  const int nblocks = (NTOT + TPB - 1) / TPB;    // 391
  WOS_72842645340328_kernel<<<dim3(nblocks), dim3(TPB), 0, stream>>>(
      x, weight, bias, mask, out);
}